// VaDEClusterer_68281390072117
// MI455X (gfx1250) — compile-verified
//
#include <hip/hip_runtime.h>

typedef float v2f __attribute__((ext_vector_type(2)));
typedef float v8f __attribute__((ext_vector_type(8)));

// D = A(16x4,f32) * B(4x16,f32) + C(16x16,f32)   -- CDNA5 V_WMMA_F32_16X16X4_F32
#define WMMA_F32(A, B, C) \
  __builtin_amdgcn_wmma_f32_16x16x4_f32(false, (A), false, (B), (short)0, (C), false, false)

constexpr int   BROWS  = 16384;
constexpr int   DDIM   = 256;
constexpr int   KDIM   = 64;
constexpr float LOG2PI = 1.8378770664093454f;

__device__ __forceinline__ float clip20(float x) {
  return fminf(fmaxf(x, -20.0f), 20.0f);
}

__device__ __forceinline__ float block_reduce256(float v, float* sm) {
  const int t = threadIdx.x;
  __syncthreads();
  sm[t] = v;
  __syncthreads();
  for (int s = 128; s > 0; s >>= 1) {
    if (t < s) sm[t] += sm[t + s];
    __syncthreads();
  }
  return sm[0];
}

// ---------------------------------------------------------------------------
// K-side precompute: weight matrices (d-pair interleaved for b64 B-fragment
// loads) and per-cluster constants.
//   U[d][k] = -c * e^{-ls}          (pairs with z)
//   W[d][k] = 0.5 * e^{-ls}         (pairs with z^2)
//   V[d][k] = -2c / clip(e^{ls})    (pairs with mu)
//   S[d][k] = 1 / clip(e^{ls})      (pairs with e^{lv} + mu^2)
//   Kc1[k]  = pi_k - 0.5*D*log2pi - Sum(ls) - Sum(c^2 * W)   (logit = Kc1 - GEMM)
//   Kc2[k]  = D*log2pi + Sum(ls) + Sum(c^2 * S)              (inner = Kc2 + GEMM)
// ---------------------------------------------------------------------------
__global__ __launch_bounds__(256) void vade_prek(
    const float* __restrict__ centroids, const float* __restrict__ log_sigma,
    const float* __restrict__ pi_logits,
    float* __restrict__ U, float* __restrict__ W,
    float* __restrict__ V, float* __restrict__ S,
    float* __restrict__ Kc1, float* __restrict__ Kc2) {
  __shared__ float sm[256];
  const int k = blockIdx.x;
  const int d = threadIdx.x;

  const float ls   = log_sigma[k * DDIM + d];
  const float c    = centroids[k * DDIM + d];
  const float eneg = expf(-ls);                       // full-accuracy here (tiny kernel)
  const float w    = 0.5f * eneg;
  const float u    = -c * eneg;
  const float es   = expf(ls);
  const float sig  = fminf(fmaxf(es, 0.0f), 20.0f);   // MIN_SIGMA=0, MAX_SIGMA=20
  const float invs = 1.0f / sig;
  const float vm   = -2.0f * c * invs;

  // pair-interleaved layout: float2 P[d>>1][k] = { X[d_even][k], X[d_even+1][k] }
  const int p = (d >> 1) * (2 * KDIM) + 2 * k + (d & 1);
  U[p] = u;  W[p] = w;  V[p] = vm;  S[p] = invs;

  const float sls = block_reduce256(ls, sm);
  const float cw  = block_reduce256(c * c * w, sm);
  const float cs  = block_reduce256(c * c * invs, sm);
  if (d == 0) {
    Kc1[k] = pi_logits[k] - 0.5f * (float)DDIM * LOG2PI - sls - cw;
    Kc2[k] = (float)DDIM * LOG2PI + sls + cs;
  }
}

// ---------------------------------------------------------------------------
// Main fused kernel: per wave = 16 B-rows x all 64 clusters.
// Pass 1 (WMMA): E-step logits -> in-wave softmax -> gamma (stored).
// Pass 2 (WMMA): inner -> loss terms, qentropy folded in via Sum(logvar).
// ---------------------------------------------------------------------------
__global__ __launch_bounds__(128) void vade_main(
    const float* __restrict__ mu, const float* __restrict__ logvar,
    const float* __restrict__ z, const float* __restrict__ pi_logits,
    const float* __restrict__ Uw, const float* __restrict__ Ww,
    const float* __restrict__ Vw, const float* __restrict__ Sw,
    const float* __restrict__ Kc1, const float* __restrict__ Kc2,
    float* __restrict__ gamma_out, float* __restrict__ partials) {
  __shared__ float red[4];
  const int lane = threadIdx.x & 31;
  const int wave = threadIdx.x >> 5;
  const int g    = lane >> 4;   // lane half-group (selects depth rows 2g,2g+1)
  const int n    = lane & 15;   // N column within a 16-wide tile / A row
  const int rb   = (blockIdx.x * 4 + wave) * 16;

  const v2f* Uv = (const v2f*)Uw;
  const v2f* Wv = (const v2f*)Ww;
  const v2f* Vv = (const v2f*)Vw;
  const v2f* Sv = (const v2f*)Sw;

  const float* zrow  = z      + (size_t)(rb + n) * DDIM + 2 * g;
  const float* murow = mu     + (size_t)(rb + n) * DDIM + 2 * g;
  const float* lvrow = logvar + (size_t)(rb + n) * DDIM + 2 * g;

  v8f zero = {};
  v8f a0 = zero, a1 = zero, a2 = zero, a3 = zero;

  // ---- Pass 1: S1[b,k] = Sum_d z*U + z^2*W  -> logits = Kc1[k] - S1 ----
#pragma unroll 2
  for (int dc = 0; dc < DDIM / 4; ++dc) {
    const int wb = (2 * dc + g) * KDIM + n;
    // batch the whole chunk's loads first (one clause), then the 8 WMMAs
    v2f zr = *(const v2f*)(zrow + 4 * dc);
    v2f u0 = Uv[wb +  0], u1 = Uv[wb + 16], u2 = Uv[wb + 32], u3 = Uv[wb + 48];
    v2f w0 = Wv[wb +  0], w1 = Wv[wb + 16], w2 = Wv[wb + 32], w3 = Wv[wb + 48];
    zr.x = clip20(zr.x);  zr.y = clip20(zr.y);
    v2f zs = zr * zr;
    a0 = WMMA_F32(zr, u0, a0);  a1 = WMMA_F32(zr, u1, a1);
    a2 = WMMA_F32(zr, u2, a2);  a3 = WMMA_F32(zr, u3, a3);
    a0 = WMMA_F32(zs, w0, a0);  a1 = WMMA_F32(zs, w1, a1);
    a2 = WMMA_F32(zs, w2, a2);  a3 = WMMA_F32(zs, w3, a3);
  }

  float kc1v[4], plv[4];
#pragma unroll
  for (int t = 0; t < 4; ++t) { kc1v[t] = Kc1[16 * t + n]; plv[t] = pi_logits[16 * t + n]; }

  v8f lg[4];
  {
    v8f tmp[4] = {a0, a1, a2, a3};
#pragma unroll
    for (int t = 0; t < 4; ++t)
#pragma unroll
      for (int v = 0; v < 8; ++v) lg[t][v] = kc1v[t] - tmp[t][v];
  }

  // ---- in-wave softmax over K=64 (row = v + 8g, cols across 16 lanes x 4 tiles) ----
  v8f  ga[4];
  float lacc = 0.0f;
#pragma unroll
  for (int v = 0; v < 8; ++v) {
    float mx = fmaxf(fmaxf(lg[0][v], lg[1][v]), fmaxf(lg[2][v], lg[3][v]));
#pragma unroll
    for (int off = 1; off < 16; off <<= 1) mx = fmaxf(mx, __shfl_xor(mx, off, 32));
    float s = 0.0f;
#pragma unroll
    for (int t = 0; t < 4; ++t) { float e = __expf(lg[t][v] - mx); ga[t][v] = e; s += e; }
#pragma unroll
    for (int off = 1; off < 16; off <<= 1) s += __shfl_xor(s, off, 32);
    const float inv = 1.0f / s;
#pragma unroll
    for (int t = 0; t < 4; ++t) {
      float gv = ga[t][v] * inv;
      ga[t][v] = gv;
      gamma_out[(size_t)(rb + v + 8 * g) * KDIM + 16 * t + n] = gv;
      lacc += gv * (__logf(gv + 1e-10f) - plv[t]);   // logqcx + logpc
    }
  }

  // ---- Pass 2: S2[b,k] = Sum_d mu*V + (e^lv + mu^2)*S -> inner = Kc2[k] + S2 ----
  a0 = zero; a1 = zero; a2 = zero; a3 = zero;
  float lvsum = 0.0f;
#pragma unroll 2
  for (int dc = 0; dc < DDIM / 4; ++dc) {
    const int wb = (2 * dc + g) * KDIM + n;
    v2f mr = *(const v2f*)(murow + 4 * dc);
    v2f lv = *(const v2f*)(lvrow + 4 * dc);
    v2f v0 = Vv[wb +  0], v1 = Vv[wb + 16], v2 = Vv[wb + 32], v3 = Vv[wb + 48];
    v2f s0 = Sv[wb +  0], s1 = Sv[wb + 16], s2 = Sv[wb + 32], s3 = Sv[wb + 48];
    mr.x = clip20(mr.x);  mr.y = clip20(mr.y);
    lv.x = clip20(lv.x);  lv.y = clip20(lv.y);
    lvsum += lv.x + lv.y;
    v2f fb;
    fb.x = __expf(lv.x) + mr.x * mr.x;
    fb.y = __expf(lv.y) + mr.y * mr.y;
    a0 = WMMA_F32(mr, v0, a0);  a1 = WMMA_F32(mr, v1, a1);
    a2 = WMMA_F32(mr, v2, a2);  a3 = WMMA_F32(mr, v3, a3);
    a0 = WMMA_F32(fb, s0, a0);  a1 = WMMA_F32(fb, s1, a1);
    a2 = WMMA_F32(fb, s2, a2);  a3 = WMMA_F32(fb, s3, a3);
  }

  float kc2v[4];
#pragma unroll
  for (int t = 0; t < 4; ++t) kc2v[t] = Kc2[16 * t + n];
  {
    v8f tmp[4] = {a0, a1, a2, a3};
#pragma unroll
    for (int t = 0; t < 4; ++t)
#pragma unroll
      for (int v = 0; v < 8; ++v)
        lacc += 0.5f * ga[t][v] * (kc2v[t] + tmp[t][v]);   // logpzc
  }
  lacc -= 0.5f * lvsum;   // logvar part of qentropy

  // wave reduce (fixed order -> deterministic)
#pragma unroll
  for (int off = 1; off < 32; off <<= 1) lacc += __shfl_xor(lacc, off, 32);
  if (lane == 0)
    red[wave] = lacc + (-0.5f * 16.0f * (float)DDIM * (1.0f + LOG2PI));  // qentropy const, 16 rows
  __syncthreads();
  if (threadIdx.x == 0)
    partials[blockIdx.x] = red[0] + red[1] + red[2] + red[3];
}

__global__ __launch_bounds__(256) void vade_finish(const float* __restrict__ partials,
                                                   int np, float* __restrict__ loss_out) {
  __shared__ float sm[256];
  float a = 0.0f;
  for (int i = threadIdx.x; i < np; i += 256) a += partials[i];
  sm[threadIdx.x] = a;
  __syncthreads();
  for (int s = 128; s > 0; s >>= 1) {
    if (threadIdx.x < s) sm[threadIdx.x] += sm[threadIdx.x + s];
    __syncthreads();
  }
  if (threadIdx.x == 0) loss_out[0] = sm[0] / (float)BROWS;
}

extern "C" void kernel_launch(void* const* d_in, const int* in_sizes, int n_in,
                              void* d_out, int out_size, void* d_ws, size_t ws_size,
                              hipStream_t stream) {
  const float* mu        = (const float*)d_in[0];
  const float* logvar    = (const float*)d_in[1];
  const float* z         = (const float*)d_in[2];
  const float* centroids = (const float*)d_in[3];
  const float* log_sigma = (const float*)d_in[4];
  const float* pi_logits = (const float*)d_in[5];
  float* out = (float*)d_out;

  float* ws  = (float*)d_ws;          // ~260 KB used
  float* U   = ws;
  float* W   = U + DDIM * KDIM;
  float* V   = W + DDIM * KDIM;
  float* S   = V + DDIM * KDIM;
  float* Kc1 = S + DDIM * KDIM;
  float* Kc2 = Kc1 + KDIM;
  float* partials = Kc2 + KDIM;

  vade_prek<<<KDIM, DDIM, 0, stream>>>(centroids, log_sigma, pi_logits,
                                       U, W, V, S, Kc1, Kc2);
  const int nblocks = BROWS / 64;     // 4 waves/block * 16 rows/wave
  vade_main<<<nblocks, 128, 0, stream>>>(mu, logvar, z, pi_logits,
                                         U, W, V, S, Kc1, Kc2,
                                         out, partials);
  vade_finish<<<1, 256, 0, stream>>>(partials, nblocks, out + (size_t)BROWS * KDIM);
}